// WeightedSAGEConv_16492674417005
// MI455X (gfx1250) — compile-verified
//
#include <hip/hip_runtime.h>

// WeightedSAGEConv for MI455X (gfx1250).
//
// Roofline: all hot state (x 20.5MB, agg 20.5MB, W 131KB, edge idx 5MB) is
// L2-resident (192MB L2), so the random gather + 82M f32 atomics run at L2
// speed; HBM sees only ~70MB compulsory traffic (~10us @ 23.3TB/s). The
// 2.6 GFLOP projection GEMM uses V_WMMA_F32_16X16X4_F32 (full f32 precision,
// matching reference numerics; compute is nowhere near the bottleneck).

typedef __attribute__((ext_vector_type(2))) float v2f;
typedef __attribute__((ext_vector_type(8))) float v8f;

#define DFEAT   128
#define DKTOT   256   // concat(x, agg) feature width
#define LDS_PITCH 257 // 16x256 tile, +1 pitch => conflict-free across 64 banks

// ---------------------------------------------------------------------------
// 0) zero workspace accumulators (harness poisons ws with 0xAA)
__global__ void zero_ws_kernel(float* __restrict__ p, int n) {
    int i = blockIdx.x * blockDim.x + threadIdx.x;
    if (i < n) p[i] = 0.0f;
}

// ---------------------------------------------------------------------------
// 1) per-destination weight sums: w_sum[dst] += w   (640K atomics, L2-resident)
__global__ void wsum_kernel(const int* __restrict__ dst,
                            const float* __restrict__ ew,
                            float* __restrict__ wsum, int E) {
    int e = blockIdx.x * blockDim.x + threadIdx.x;
    if (e < E) atomicAdd(&wsum[dst[e]], ew[e]);
}

// ---------------------------------------------------------------------------
// 2) repack W [128][256] into WMMA-B-fragment order:
//    Wt2[(k>>1)*256 + o*2 + (k&1)] = W[o][k]
//    => each lane's B frag (K pair kk,kk+1 at column o) is one contiguous
//       8-byte load, coalesced across lanes 0..15 (global_load_b64).
__global__ void pack_W_kernel(const float* __restrict__ W,
                              float* __restrict__ Wt2) {
    int i = blockIdx.x * blockDim.x + threadIdx.x;   // over 128*256
    if (i >= DFEAT * DKTOT) return;
    int o = i & (DFEAT - 1);   // 0..127 (consecutive across lanes)
    int k = i >> 7;            // 0..255
    Wt2[(k >> 1) * (2 * DFEAT) + o * 2 + (k & 1)] = W[o * DKTOT + k];
}

// ---------------------------------------------------------------------------
// 3) weighted scatter-add: agg[dst] += (w / clip(wsum[dst])) * x[src]
//    32 lanes per edge, float4 per lane -> gather is a coalesced 512B row,
//    atomics land in L2.
__global__ void scatter_kernel(const int* __restrict__ src,
                               const int* __restrict__ dst,
                               const float* __restrict__ ew,
                               const float* __restrict__ wsum,
                               const float* __restrict__ x,
                               float* __restrict__ agg, int E) {
    int gid = blockIdx.x * blockDim.x + threadIdx.x;
    int e  = gid >> 5;            // 32 threads per edge
    int d4 = (gid & 31) << 2;     // 4 features per thread
    if (e >= E) return;
    int s = src[e];
    int t = dst[e];
    float wn = ew[e] / fmaxf(wsum[t], 1e-8f);
    const float4 v = *(const float4*)(x + (size_t)s * DFEAT + d4);
    float* o = agg + (size_t)t * DFEAT + d4;
    atomicAdd(o + 0, wn * v.x);
    atomicAdd(o + 1, wn * v.y);
    atomicAdd(o + 2, wn * v.z);
    atomicAdd(o + 3, wn * v.w);
}

// ---------------------------------------------------------------------------
// 4) projection GEMM: out[n,o] = sum_k h[n,k]*Wt[k,o] + b[o], h=[x|agg].
//    Block = 256 threads = 8 wave32. Block owns a 16-node tile of h staged in
//    LDS; wave w computes the 16x16 output tile at columns [16w,16w+16) via
//    64 x v_wmma_f32_16x16x4_f32. Last tile is handled by clamping m0 so the
//    WMMA epilogue needs no per-row predication (overlapping blocks write
//    identical values -> deterministic).
__global__ __launch_bounds__(256) void gemm_wmma_kernel(
        const float* __restrict__ x, const float* __restrict__ agg,
        const float* __restrict__ Wt2, const float* __restrict__ bias,
        float* __restrict__ out, int nNodes) {
    __shared__ float hlds[16 * LDS_PITCH];

    int m0 = blockIdx.x * 16;
    if (m0 > nNodes - 16) m0 = nNodes - 16;   // clamp: no guards needed below
    const int t = threadIdx.x;

    // Cooperative load of the h tile [16 x 256] (cols 0..127 from x,
    // 128..255 from agg). Linear j => fully coalesced 16-row sweep.
    for (int i = 0; i < 16; ++i) {
        int j   = t + 256 * i;
        int row = j >> 8;          // 0..15
        int col = j & 255;         // 0..255
        size_t n = (size_t)(m0 + row);
        float v = (col < DFEAT) ? x[n * DFEAT + col]
                                : agg[n * DFEAT + (col - DFEAT)];
        hlds[row * LDS_PITCH + col] = v;
    }
    __syncthreads();

    const int lane  = t & 31;
    const int laneM = lane & 15;       // M (A) / N (B,C,D) index
    const int hi    = lane >> 4;       // half-wave select
    const int o0    = (t >> 5) * 16;   // wave's output-column tile

    // A 16x4 f32 frag: lane = M row; VGPR0/1 = K{0,1} (lo half) / K{2,3} (hi).
    // B 4x16 f32 frag: lane = N col; VGPR0/1 = K{0,1} (lo half) / K{2,3} (hi).
    const float* hrow = hlds + laneM * LDS_PITCH;
    const float* bcol = Wt2 + (size_t)(o0 + laneM) * 2;  // + (kk>>1)*256
    v8f c = {};
    for (int k = 0; k < DKTOT; k += 4) {
        const int kk = k + 2 * hi;     // always even
        v2f a;
        a.x = hrow[kk + 0];
        a.y = hrow[kk + 1];
        const v2f b = *(const v2f*)(bcol + (size_t)(kk >> 1) * (2 * DFEAT));
        c = __builtin_amdgcn_wmma_f32_16x16x4_f32(
                /*neg_a=*/false, a, /*neg_b=*/false, b,
                /*c_mod=*/(short)0, c, /*reuse_a=*/false, /*reuse_b=*/false);
    }

    // C/D 16x16 f32 layout: VGPR r -> M = r + 8*hi, N = laneM.
    const float bb = bias[o0 + laneM];
    float* orow = out + (size_t)(m0 + 8 * hi) * DFEAT + o0 + laneM;
    for (int r = 0; r < 8; ++r) {
        orow[(size_t)r * DFEAT] = c[r] + bb;
    }
}

// ---------------------------------------------------------------------------
extern "C" void kernel_launch(void* const* d_in, const int* in_sizes, int n_in,
                              void* d_out, int out_size, void* d_ws, size_t ws_size,
                              hipStream_t stream) {
    const float* x  = (const float*)d_in[0];
    const int*   ei = (const int*)d_in[1];   // [2, E]
    const float* ew = (const float*)d_in[2];
    const float* W  = (const float*)d_in[3]; // [128, 256]
    const float* b  = (const float*)d_in[4]; // [128]

    const int nNodes = in_sizes[0] / DFEAT;  // 40000
    const int E      = in_sizes[2];          // 640000
    const int* src = ei;
    const int* dst = ei + E;

    // workspace layout: w_sum | agg | Wt2
    float* wsum = (float*)d_ws;                              // nNodes
    float* agg  = wsum + nNodes;                             // nNodes * 128
    float* Wt2  = agg + (size_t)nNodes * DFEAT;              // 256 * 128
    float* out  = (float*)d_out;

    const int zn = nNodes + nNodes * DFEAT;
    zero_ws_kernel<<<(zn + 255) / 256, 256, 0, stream>>>(wsum, zn);

    wsum_kernel<<<(E + 255) / 256, 256, 0, stream>>>(dst, ew, wsum, E);

    pack_W_kernel<<<(DFEAT * DKTOT + 255) / 256, 256, 0, stream>>>(W, Wt2);

    const int sth = E * 32;  // 32 lanes per edge
    scatter_kernel<<<(sth + 255) / 256, 256, 0, stream>>>(src, dst, ew, wsum,
                                                          x, agg, E);

    gemm_wmma_kernel<<<(nNodes + 15) / 16, 256, 0, stream>>>(x, agg, Wt2, b,
                                                             out, nNodes);
}